// TanhRT_45406394253467
// MI455X (gfx1250) — compile-verified
//
#include <hip/hip_runtime.h>
#include <stdint.h>

// out[b,n,m,f] = e0 + e1 * tanh((z - e2) * e3), params gathered per (b,f).
// Pure HBM-streaming kernel: 256MB in + 256MB out => ~22us floor @ 23.3TB/s.
// Strategy: b128 NT loads/stores, params held in registers, v_tanh_f32 on the
// trans unit, mask row staged via gfx1250 async load-to-LDS.

typedef float f32x4 __attribute__((ext_vector_type(4)));

#define B_ 16
#define N_ 8
#define M_ 1024
#define F_ 512
#define ROWS_TOTAL (N_ * M_)              // rows per batch = 8192
#define THREADS 256
#define CHUNKS 128                        // blocks per batch
#define ROWS_PER_CHUNK (ROWS_TOTAL / CHUNKS)   // 64
#define ROWS_PER_ITER ((THREADS * 4) / F_)     // 2 rows per block iteration
#define F4 (F_ / 4)                       // 128 float4 per row

#define AS1 __attribute__((address_space(1)))
#define AS3 __attribute__((address_space(3)))

__device__ __forceinline__ float fast_tanh(float x) {
#if __has_builtin(__builtin_amdgcn_tanhf)
  return __builtin_amdgcn_tanhf(x);
#else
  float r;
  // CDNA5 TRANS op; embed v_nop to honor the 1-op trans-result hazard.
  asm volatile("v_tanh_f32 %0, %1\n\tv_nop" : "=v"(r) : "v"(x));
  return r;
#endif
}

__global__ __launch_bounds__(THREADS) void tanh_fault_kernel(
    const float* __restrict__ z,
    const int* __restrict__ mask,       // [B,F], values 0..12
    const float* __restrict__ eta,      // [13,4] = 16B rows (float4 aligned)
    float* __restrict__ out)
{
  __shared__ int   s_mask[F_];
  __shared__ f32x4 s_e[F_];             // (e0,e1,e2,e3) per f

  const int b     = blockIdx.x / CHUNKS;
  const int chunk = blockIdx.x % CHUNKS;
  const int tid   = threadIdx.x;

  // ---- Stage mask[b, :] into LDS (gfx1250 async-to-LDS; ASYNCcnt path) ----
  const int* gm = mask + b * F_;
#if __has_builtin(__builtin_amdgcn_global_load_async_to_lds_b32)
  {
    AS1 int* g0 = (AS1 int*)(uintptr_t)(gm + tid);
    AS1 int* g1 = (AS1 int*)(uintptr_t)(gm + tid + THREADS);
    AS3 int* l0 = (AS3 int*)(unsigned)(uintptr_t)&s_mask[tid];
    AS3 int* l1 = (AS3 int*)(unsigned)(uintptr_t)&s_mask[tid + THREADS];
    __builtin_amdgcn_global_load_async_to_lds_b32(g0, l0, 0, 0);
    __builtin_amdgcn_global_load_async_to_lds_b32(g1, l1, 0, 0);
#if __has_builtin(__builtin_amdgcn_s_wait_asynccnt)
    __builtin_amdgcn_s_wait_asynccnt(0);
#else
    asm volatile("s_wait_asynccnt 0" ::: "memory");
#endif
  }
#else
  s_mask[tid]           = gm[tid];
  s_mask[tid + THREADS] = gm[tid + THREADS];
#endif
  __syncthreads();

  // ---- Expand mask -> per-f eta float4 in LDS ----
  {
    int m0 = s_mask[tid];
    int m1 = s_mask[tid + THREADS];
    s_e[tid]           = *(const f32x4*)(eta + m0 * 4);
    s_e[tid + THREADS] = *(const f32x4*)(eta + m1 * 4);
  }
  __syncthreads();

  // ---- Pin this thread's 4 eta vectors (f = 4*f4 .. 4*f4+3) in registers ----
  const int f4   = tid & (F4 - 1);      // float4 column within a row
  const int rsub = tid >> 7;            // 0/1: which row of the pair
  const f32x4 ea = s_e[4 * f4 + 0];
  const f32x4 eb = s_e[4 * f4 + 1];
  const f32x4 ec = s_e[4 * f4 + 2];
  const f32x4 ed = s_e[4 * f4 + 3];

  // ---- Stream 64 rows: NT b128 load -> tanh math -> NT b128 store ----
  const f32x4* __restrict__ zi = (const f32x4*)z;
  f32x4* __restrict__       oo = (f32x4*)out;
  const int rowBase = b * ROWS_TOTAL + chunk * ROWS_PER_CHUNK;

#pragma unroll 4
  for (int rr = rsub; rr < ROWS_PER_CHUNK; rr += ROWS_PER_ITER) {
    const int idx = (rowBase + rr) * F4 + f4;     // max 16M float4 -> fits int
    f32x4 zv = __builtin_nontemporal_load(zi + idx);
    f32x4 r;
    r.x = ea.x + ea.y * fast_tanh((zv.x - ea.z) * ea.w);
    r.y = eb.x + eb.y * fast_tanh((zv.y - eb.z) * eb.w);
    r.z = ec.x + ec.y * fast_tanh((zv.z - ec.z) * ec.w);
    r.w = ed.x + ed.y * fast_tanh((zv.w - ed.z) * ed.w);
    __builtin_nontemporal_store(r, oo + idx);
  }
}

extern "C" void kernel_launch(void* const* d_in, const int* in_sizes, int n_in,
                              void* d_out, int out_size, void* d_ws, size_t ws_size,
                              hipStream_t stream) {
  (void)in_sizes; (void)n_in; (void)d_ws; (void)ws_size; (void)out_size;
  const float* z    = (const float*)d_in[0];
  const int*   mask = (const int*)d_in[1];   // harness contract: int -> const int*
  const float* eta  = (const float*)d_in[2];
  float*       out  = (float*)d_out;

  dim3 grid(B_ * CHUNKS);                     // 2048 blocks
  dim3 block(THREADS);                        // 8 waves/block (wave32)
  tanh_fault_kernel<<<grid, block, 0, stream>>>(z, mask, eta, out);
}